// DGCNN_encoder_59528246723043
// MI455X (gfx1250) — compile-verified
//
#include <hip/hip_runtime.h>

typedef float v2f __attribute__((ext_vector_type(2)));
typedef float v8f __attribute__((ext_vector_type(8)));

#define KNN   27
#define NPTS  256
#define NB    4            // B*T flat batches
#define NCOLS (NPTS * KNN) // 6912 = 54*128
#define CCAT  8160
#define LATENT 256

// ---------------- prep: x (b,n,3) -> xprep (b,3,n) ----------------
__global__ void prepx_kernel(const float* __restrict__ x, float* __restrict__ xp) {
    int t = blockIdx.x * blockDim.x + threadIdx.x;
    if (t >= NB * 3 * NPTS) return;
    int b = t / (3 * NPTS);
    int r = t - b * 3 * NPTS;
    int c = r / NPTS;
    int n = r - c * NPTS;
    xp[t] = x[((size_t)b * NPTS + n) * 3 + c];
}

// ---------------- KNN: top-27 by squared distance ----------------
__global__ __launch_bounds__(256) void knn_kernel(const float* __restrict__ x,
                                                  int* __restrict__ idx) {
    __shared__ float px[NPTS], py[NPTS], pz[NPTS];
    int b = blockIdx.x, t = threadIdx.x;
    const float* xb = x + (size_t)b * NPTS * 3;
    px[t] = xb[t * 3 + 0];
    py[t] = xb[t * 3 + 1];
    pz[t] = xb[t * 3 + 2];
    __syncthreads();
    float cx = px[t], cy = py[t], cz = pz[t];
    float bd[KNN];
    int   bi[KNN];
    for (int i = 0; i < KNN; i++) { bd[i] = 3.4e38f; bi[i] = -1; }
    for (int j = 0; j < NPTS; j++) {
        float dx = px[j] - cx, dy = py[j] - cy, dz = pz[j] - cz;
        float d = dx * dx + dy * dy + dz * dz;
        if (d < bd[KNN - 1]) {           // strict: ties keep lower index (top_k semantics)
            int p = KNN - 1;
            while (p > 0 && d < bd[p - 1]) {
                bd[p] = bd[p - 1]; bi[p] = bi[p - 1]; p--;
            }
            bd[p] = d; bi[p] = j;
        }
    }
    int* o = idx + ((size_t)b * NPTS + t) * KNN;
    for (int i = 0; i < KNN; i++) o[i] = bi[i];
}

// ---------------- fused gather + GEMM (f32 WMMA 16x16x4) ----------------
// out[b][o][col] = sum_c W[o][c] * feat[b][c][col]
// mode 1 (graph): col=(n,kk); c<Cin: prev[c][idx[n,kk]]-prev[c][n]; else prev[c-Cin][n]
// mode 0 (plain): feat[c][col] = prev[c][col]
#define BM 128
#define BN 128
#define BK 16
#define LDK (BK + 2)   // pad: 18 floats = 72B rows -> 8B-aligned even-k b64 frags

__global__ __launch_bounds__(256) void gemm_kernel(
    const float* __restrict__ W, const float* __restrict__ prev,
    const int* __restrict__ idx, float* __restrict__ out,
    int Cout, int Cin2, int Cin, int prevBatchStride, int Ncols, int mode)
{
    // double-buffered tiles; K-fastest so fragments are single ds_load_b64
    __shared__ __align__(16) float Wlds [2][BM][LDK];
    __shared__ __align__(16) float FldsT[2][BN][LDK];

    int b   = blockIdx.z;
    int N0  = blockIdx.x * BN;
    int M0  = blockIdx.y * BM;
    int tid = threadIdx.x;
    int lane = tid & 31, wid = tid >> 5;
    int wm = (wid & 3) * 32;   // 4 waves along M (32 rows each)
    int wn = (wid >> 2) * 64;  // 2 waves along N (64 cols each)
    int l15 = lane & 15, lh = lane >> 4;

    v8f acc[2][4] = {};

    const float* prevb = prev + (size_t)b * prevBatchStride;
    int nchunks = (Cin2 + BK - 1) / BK;

    auto stage = [&](int kc, int sb) {
        int kbase = kc * BK;
        bool fullW = (M0 + BM <= Cout) && (kbase + BK <= Cin2);
        if (fullW) {
            // async DMA: global -> LDS, 1024 x b64 segments, ASYNCcnt-tracked
            #pragma unroll
            for (int r = 0; r < 4; r++) {
                int s = tid + 256 * r;
                int m = s >> 3, c2 = (s & 7) << 1;
                const float* gp = W + (size_t)(M0 + m) * Cin2 + (kbase + c2);
                unsigned lo = (unsigned)(uintptr_t)(&Wlds[sb][m][c2]);
                asm volatile("global_load_async_to_lds_b64 %0, %1, off"
                             :: "v"(lo), "v"(gp) : "memory");
            }
        } else {
            #pragma unroll
            for (int r = 0; r < 8; r++) {
                int e = tid + 256 * r;
                int m = e >> 4, cl = e & 15;
                int o = M0 + m, c = kbase + cl;
                Wlds[sb][m][cl] = (o < Cout && c < Cin2) ? W[(size_t)o * Cin2 + c] : 0.0f;
            }
            if (kbase + BK < Cin2) {
                int om = M0 + (tid >> 1);
                if (om < Cout)
                    __builtin_prefetch(&W[(size_t)om * Cin2 + kbase + BK], 0, 1);
            }
        }
        // feat tile 16(K) x 128(N), stored transposed [col][k]; gather on the fly
        #pragma unroll
        for (int r = 0; r < 8; r++) {
            int e = tid + 256 * r;
            int cl = e >> 7, j = e & 127;
            int c = kbase + cl;
            int col = N0 + j;
            float v = 0.0f;
            if (c < Cin2 && col < Ncols) {
                if (mode == 1) {
                    int n = col / KNN;
                    int kk = col - n * KNN;
                    if (c < Cin) {
                        int nb = idx[((size_t)b * NPTS + n) * KNN + kk];
                        v = prevb[c * NPTS + nb] - prevb[c * NPTS + n];
                    } else {
                        v = prevb[(c - Cin) * NPTS + n];
                    }
                } else {
                    v = prevb[c * NPTS + col];
                }
            }
            FldsT[sb][j][cl] = v;
        }
    };

    stage(0, 0);
    asm volatile("s_wait_asynccnt 0x0" ::: "memory");
    __syncthreads();

    for (int kc = 0; kc < nchunks; kc++) {
        int cb = kc & 1;
        if (kc + 1 < nchunks) stage(kc + 1, cb ^ 1);

        #pragma unroll
        for (int k4 = 0; k4 < BK; k4 += 4) {
            int kA = k4 + 2 * lh;  // even -> 8B aligned b64 fragment loads
            v2f a0 = *reinterpret_cast<const v2f*>(&Wlds [cb][wm + l15][kA]);
            v2f a1 = *reinterpret_cast<const v2f*>(&Wlds [cb][wm + 16 + l15][kA]);
            v2f b0 = *reinterpret_cast<const v2f*>(&FldsT[cb][wn + l15][kA]);
            v2f b1 = *reinterpret_cast<const v2f*>(&FldsT[cb][wn + 16 + l15][kA]);
            v2f b2 = *reinterpret_cast<const v2f*>(&FldsT[cb][wn + 32 + l15][kA]);
            v2f b3 = *reinterpret_cast<const v2f*>(&FldsT[cb][wn + 48 + l15][kA]);
            acc[0][0] = __builtin_amdgcn_wmma_f32_16x16x4_f32(false, a0, false, b0, (short)0, acc[0][0], false, false);
            acc[0][1] = __builtin_amdgcn_wmma_f32_16x16x4_f32(false, a0, false, b1, (short)0, acc[0][1], false, false);
            acc[0][2] = __builtin_amdgcn_wmma_f32_16x16x4_f32(false, a0, false, b2, (short)0, acc[0][2], false, false);
            acc[0][3] = __builtin_amdgcn_wmma_f32_16x16x4_f32(false, a0, false, b3, (short)0, acc[0][3], false, false);
            acc[1][0] = __builtin_amdgcn_wmma_f32_16x16x4_f32(false, a1, false, b0, (short)0, acc[1][0], false, false);
            acc[1][1] = __builtin_amdgcn_wmma_f32_16x16x4_f32(false, a1, false, b1, (short)0, acc[1][1], false, false);
            acc[1][2] = __builtin_amdgcn_wmma_f32_16x16x4_f32(false, a1, false, b2, (short)0, acc[1][2], false, false);
            acc[1][3] = __builtin_amdgcn_wmma_f32_16x16x4_f32(false, a1, false, b3, (short)0, acc[1][3], false, false);
        }
        asm volatile("s_wait_asynccnt 0x0" ::: "memory");
        __syncthreads();
    }

    // D layout: lanes 0-15: (M=v, N=lane); lanes 16-31: (M=v+8, N=lane-16)
    #pragma unroll
    for (int mi = 0; mi < 2; mi++) {
        #pragma unroll
        for (int ni = 0; ni < 4; ni++) {
            int n = N0 + wn + ni * 16 + l15;
            if (n >= Ncols) continue;
            #pragma unroll
            for (int v = 0; v < 8; v++) {
                int o = M0 + wm + mi * 16 + v + 8 * lh;
                if (o < Cout)
                    out[((size_t)b * Cout + o) * Ncols + n] = acc[mi][ni][v];
            }
        }
    }
}

// ---------------- per-channel mean / rstd over (b, cols) ----------------
__global__ __launch_bounds__(256) void stats_kernel(const float* __restrict__ h,
                                                    float* __restrict__ mean,
                                                    float* __restrict__ rstd,
                                                    int C, int Ncols) {
    int o = blockIdx.x, tid = threadIdx.x;
    int total = NB * Ncols;
    float s1 = 0.f, s2 = 0.f;
    for (int i = tid; i < total; i += 256) {
        int b = i / Ncols, col = i - b * Ncols;
        float v = h[((size_t)b * C + o) * Ncols + col];
        s1 += v; s2 += v * v;
    }
    __shared__ float r1[256], r2[256];
    r1[tid] = s1; r2[tid] = s2;
    __syncthreads();
    for (int s = 128; s > 0; s >>= 1) {
        if (tid < s) { r1[tid] += r1[tid + s]; r2[tid] += r2[tid + s]; }
        __syncthreads();
    }
    if (tid == 0) {
        float m = r1[0] / (float)total;
        float var = r2[0] / (float)total - m * m;
        mean[o] = m;
        rstd[o] = rsqrtf(var + 1e-5f);
    }
}

// ---------------- BN + LeakyReLU + max over k -> xcat slice ----------------
__global__ void bnmax_kernel(const float* __restrict__ h, const float* __restrict__ mean,
                             const float* __restrict__ rstd, const float* __restrict__ gamma,
                             const float* __restrict__ beta, float* __restrict__ xcat,
                             int Cout, int off) {
    int t = blockIdx.x * 256 + threadIdx.x;
    int tot = NB * Cout * NPTS;
    if (t >= tot) return;
    int bt = t / (Cout * NPTS);
    int r  = t - bt * (Cout * NPTS);
    int o  = r / NPTS;
    int n  = r - o * NPTS;
    const float* hp = h + ((size_t)bt * Cout + o) * NCOLS + (size_t)n * KNN;
    float sc = rstd[o] * gamma[o];
    float sh = beta[o] - mean[o] * sc;
    float m = -3.4e38f;
    for (int kk = 0; kk < KNN; kk++) {
        float y = hp[kk] * sc + sh;
        y = (y >= 0.f) ? y : 0.2f * y;
        m = fmaxf(m, y);
    }
    xcat[((size_t)bt * CCAT + off + o) * NPTS + n] = m;
}

// ---------------- temporal pool half: tp = mean over T of batch-0 ----------------
__global__ void tpool_kernel(float* __restrict__ xcat, int Cout, int off) {
    int t = blockIdx.x * 256 + threadIdx.x;
    if (t >= Cout * NPTS) return;
    int o = t / NPTS, n = t - o * NPTS;
    float a  = xcat[((size_t)0 * CCAT + off + o) * NPTS + n];
    float bv = xcat[((size_t)1 * CCAT + off + o) * NPTS + n];
    float tp = 0.5f * (a + bv);
    for (int bt = 0; bt < NB; bt++)
        xcat[((size_t)bt * CCAT + off + Cout + o) * NPTS + n] = tp;
}

// ---------------- final BN + LeakyReLU -> d_out ----------------
__global__ void final_kernel(const float* __restrict__ ypre, const float* __restrict__ mean,
                             const float* __restrict__ rstd, const float* __restrict__ gamma,
                             const float* __restrict__ beta, float* __restrict__ out) {
    int t = blockIdx.x * 256 + threadIdx.x;
    if (t >= NB * LATENT * NPTS) return;
    int o = (t / NPTS) & (LATENT - 1);
    float sc = rstd[o] * gamma[o];
    float sh = beta[o] - mean[o] * sc;
    float y = ypre[t] * sc + sh;
    out[t] = (y >= 0.f) ? y : 0.2f * y;
}

// ---------------- host ----------------
extern "C" void kernel_launch(void* const* d_in, const int* in_sizes, int n_in,
                              void* d_out, int out_size, void* d_ws, size_t ws_size,
                              hipStream_t stream) {
    const float* x = (const float*)d_in[0];
    const float* Ws[5] = {(const float*)d_in[1], (const float*)d_in[4], (const float*)d_in[7],
                          (const float*)d_in[10], (const float*)d_in[13]};
    const float* Gs[5] = {(const float*)d_in[2], (const float*)d_in[5], (const float*)d_in[8],
                          (const float*)d_in[11], (const float*)d_in[14]};
    const float* Bs[5] = {(const float*)d_in[3], (const float*)d_in[6], (const float*)d_in[9],
                          (const float*)d_in[12], (const float*)d_in[15]};

    char* ws = (char*)d_ws;
    size_t OFF_IDX   = 0;                               // 4*256*27 int
    size_t OFF_XPREP = OFF_IDX   + 110592;              // 4*3*256 f32
    size_t OFF_MEAN  = OFF_XPREP + 12288;               // 3072 f32
    size_t OFF_RSTD  = OFF_MEAN  + 12288;               // 3072 f32
    size_t OFF_YPRE  = OFF_RSTD  + 12288;               // 4*256*256 f32
    size_t OFF_XCAT  = OFF_YPRE  + 1048576;             // 4*8160*256 f32
    size_t OFF_H     = OFF_XCAT  + 33423360;            // 4*3072*6912 f32 (339.7 MB)

    int*   idx   = (int*)(ws + OFF_IDX);
    float* xprep = (float*)(ws + OFF_XPREP);
    float* meanb = (float*)(ws + OFF_MEAN);
    float* rstdb = (float*)(ws + OFF_RSTD);
    float* ypre  = (float*)(ws + OFF_YPRE);
    float* xcat  = (float*)(ws + OFF_XCAT);
    float* hbuf  = (float*)(ws + OFF_H);

    prepx_kernel<<<(NB * 3 * NPTS + 255) / 256, 256, 0, stream>>>(x, xprep);
    knn_kernel<<<NB, 256, 0, stream>>>(x, idx);

    int Cprevs[4] = {3, 96, 384, 1536};
    int Couts[4]  = {48, 192, 768, 3072};
    int offs[4]   = {0, 96, 480, 2016};
    const float* prevPtrs[4] = {xprep, xcat, xcat + 96 * NPTS, xcat + 480 * NPTS};
    int pbs[4] = {3 * NPTS, CCAT * NPTS, CCAT * NPTS, CCAT * NPTS};

    for (int l = 0; l < 4; l++) {
        int Cin  = Cprevs[l];
        int Cin2 = 2 * Cin;
        int Cout = Couts[l];
        dim3 grid((NCOLS + BN - 1) / BN, (Cout + BM - 1) / BM, NB);
        gemm_kernel<<<grid, 256, 0, stream>>>(Ws[l], prevPtrs[l], idx, hbuf,
                                              Cout, Cin2, Cin, pbs[l], NCOLS, 1);
        stats_kernel<<<Cout, 256, 0, stream>>>(hbuf, meanb, rstdb, Cout, NCOLS);
        bnmax_kernel<<<(NB * Cout * NPTS + 255) / 256, 256, 0, stream>>>(
            hbuf, meanb, rstdb, Gs[l], Bs[l], xcat, Cout, offs[l]);
        tpool_kernel<<<(Cout * NPTS + 255) / 256, 256, 0, stream>>>(xcat, Cout, offs[l]);
    }

    // final projection: (256 x 8160) @ xcat(8160 x 256) per batch
    {
        dim3 grid((NPTS + BN - 1) / BN, (LATENT + BM - 1) / BM, NB);
        gemm_kernel<<<grid, 256, 0, stream>>>(Ws[4], xcat, idx, ypre,
                                              LATENT, CCAT, 0, CCAT * NPTS, NPTS, 0);
        stats_kernel<<<LATENT, 256, 0, stream>>>(ypre, meanb, rstdb, LATENT, NPTS);
        final_kernel<<<(NB * LATENT * NPTS + 255) / 256, 256, 0, stream>>>(
            ypre, meanb, rstdb, Gs[4], Bs[4], (float*)d_out);
    }
    (void)in_sizes; (void)n_in; (void)out_size; (void)ws_size;
}